// GCF_1228360647041
// MI455X (gfx1250) — compile-verified
//
#include <hip/hip_runtime.h>
#include <hip/hip_bf16.h>

typedef __attribute__((ext_vector_type(2))) float v2f;
typedef __attribute__((ext_vector_type(8))) float v8f;

#define D 128

// ---------------- degree / normalization (computed ONCE; layer-invariant) ---
__global__ void k_deg_init(int* __restrict__ deg, int n) {
  int i = blockIdx.x * blockDim.x + threadIdx.x;
  if (i < n) deg[i] = 1;  // self-loop contributes 1 to every node
}

__global__ void k_deg_count(const int* __restrict__ ui, const int* __restrict__ ii,
                            int* __restrict__ deg, int B) {
  int e = blockIdx.x * blockDim.x + threadIdx.x;
  if (e >= B) return;
  atomicAdd(&deg[ii[e]], 1);       // dst = item
  atomicAdd(&deg[ui[e] + B], 1);   // dst = user shifted by n/2 == B
}

__global__ void k_dinv(const int* __restrict__ deg, float* __restrict__ dinv, int n) {
  int i = blockIdx.x * blockDim.x + threadIdx.x;
  if (i < n) dinv[i] = rsqrtf((float)deg[i]);  // deg >= 1 always
}

// ---------------- initial embedding gather:  X = [user_emb ; item_emb] ------
__global__ void k_gather(const int* __restrict__ ui, const int* __restrict__ ii,
                         const float* __restrict__ utab, const float* __restrict__ itab,
                         float* __restrict__ X, int B) {
  int t = blockIdx.x * blockDim.x + threadIdx.x;
  int row = t >> 5, c = t & 31;          // 32 lanes * float4 = 128 floats/row
  if (row >= 2 * B) return;
  const float4* src = (row < B)
      ? (const float4*)(utab + (size_t)ui[row] * D)
      : (const float4*)(itab + (size_t)ii[row - B] * D);
  ((float4*)(X + (size_t)row * D))[c] = src[c];
}

// ---------------- H = X @ W  via V_WMMA_F32_16X16X4_F32 ---------------------
// Block = 256 threads = 8 waves; each wave computes 16 rows x 128 cols.
// LDS holds W swizzled as K-pairs: lw[kp*128 + c] = {W[2kp][c], W[2kp+1][c]}
// so a single 8-byte LDS read yields a lane's B fragment.
__global__ void __launch_bounds__(256)
k_gemm_wmma(const float* __restrict__ X, const float* __restrict__ W,
            float* __restrict__ H, int nrows) {
  __shared__ float2 lw[64 * 128];  // 64 KB (of 320 KB/WGP)
  const int tid = threadIdx.x;
  for (int i = tid; i < 64 * 128; i += 256) {
    int kp = i >> 7, c = i & 127;
    float2 v;
    v.x = W[(2 * kp) * D + c];
    v.y = W[(2 * kp + 1) * D + c];
    lw[i] = v;
  }
  __syncthreads();

  const int lane = tid & 31;
  const int wv   = tid >> 5;
  const int r0   = blockIdx.x * 128 + wv * 16;
  if (r0 >= nrows) return;  // n is a multiple of 128: no intra-block divergence

  // A fragment (16x4 f32 layout): lanes 0-15 hold K={k0,k0+1} of row M=lane,
  // lanes 16-31 hold K={k0+2,k0+3} of row M=lane-16.  Full K=128 -> 32 x v2f.
  v2f a[32];
  const float* xp = X + (size_t)(r0 + (lane & 15)) * D + ((lane >> 4) << 1);
#pragma unroll
  for (int t = 0; t < 32; ++t)
    a[t] = *(const v2f*)(xp + 4 * t);

  const int kpb   = lane >> 4;   // 0: rows {4s,4s+1}; 1: rows {4s+2,4s+3}
  const int cbase = lane & 15;

  for (int n = 0; n < 8; ++n) {  // 8 tiles of 16 output columns
    v8f acc = {};
    const float2* bp = &lw[kpb * 128 + n * 16 + cbase];
#pragma unroll
    for (int s = 0; s < 32; ++s) {           // K = 128 in steps of 4
      float2 bl = bp[s * 256];               // element (2s+kpb)*128 + col
      v2f b; b[0] = bl.x; b[1] = bl.y;
      acc = __builtin_amdgcn_wmma_f32_16x16x4_f32(
          /*neg_a=*/false, a[s], /*neg_b=*/false, b,
          /*c_mod=*/(short)0, acc, /*reuse_a=*/false, /*reuse_b=*/false);
    }
    // C/D layout: VGPR j -> (M=j, N=lane) for lanes 0-15, (M=8+j, N=lane-16)
    float* hp = H + (size_t)(r0 + (kpb ? 8 : 0)) * D + n * 16 + cbase;
#pragma unroll
    for (int j = 0; j < 8; ++j)
      hp[(size_t)j * D] = acc[j];
  }
}

// ---------------- zero the accumulator (X is dead after the GEMM) -----------
__global__ void k_zero(float4* __restrict__ p, int n4) {
  int i = blockIdx.x * blockDim.x + threadIdx.x;
  if (i < n4) { float4 z = {0.f, 0.f, 0.f, 0.f}; p[i] = z; }
}

// ---------------- edge scatter: one wave per interaction (both directions) --
__global__ void k_scatter(const float* __restrict__ H,
                          const int* __restrict__ ui, const int* __restrict__ ii,
                          const float* __restrict__ dinv,
                          float* __restrict__ acc, int B) {
  int e    = blockIdx.x * (blockDim.x >> 5) + (threadIdx.x >> 5);
  int lane = threadIdx.x & 31;
  if (e >= B) return;
  int u  = ui[e] + B;
  int it = ii[e];
  float nrm = dinv[u] * dinv[it];                 // symmetric for both dirs
  float4 hu = ((const float4*)(H + (size_t)u  * D))[lane];
  float4 hi = ((const float4*)(H + (size_t)it * D))[lane];
  float* ou = acc + (size_t)u  * D + lane * 4;
  float* oi = acc + (size_t)it * D + lane * 4;
  atomicAdd(oi + 0, nrm * hu.x);  // out[item] += norm * h[user]
  atomicAdd(oi + 1, nrm * hu.y);
  atomicAdd(oi + 2, nrm * hu.z);
  atomicAdd(oi + 3, nrm * hu.w);
  atomicAdd(ou + 0, nrm * hi.x);  // out[user] += norm * h[item]
  atomicAdd(ou + 1, nrm * hi.y);
  atomicAdd(ou + 2, nrm * hi.z);
  atomicAdd(ou + 3, nrm * hi.w);
}

// ---------------- epilogue: fused self-loop + bias + ReLU -------------------
// x = relu(acc + dinv[i]^2 * h[i] + b)
__global__ void k_finish(float* __restrict__ X, const float* __restrict__ H,
                         const float* __restrict__ dinv,
                         const float* __restrict__ bias, int n) {
  int t = blockIdx.x * blockDim.x + threadIdx.x;
  int row = t >> 5, c = t & 31;
  if (row >= n) return;
  float di = dinv[row];
  float s  = di * di;
  float4 h = ((const float4*)(H + (size_t)row * D))[c];
  float4 x = ((const float4*)(X + (size_t)row * D))[c];
  float4 b = ((const float4*)bias)[c];
  float4 r;
  r.x = fmaxf(fmaf(s, h.x, x.x) + b.x, 0.f);
  r.y = fmaxf(fmaf(s, h.y, x.y) + b.y, 0.f);
  r.z = fmaxf(fmaf(s, h.z, x.z) + b.z, 0.f);
  r.w = fmaxf(fmaf(s, h.w, x.w) + b.w, 0.f);
  ((float4*)(X + (size_t)row * D))[c] = r;
}

extern "C" void kernel_launch(void* const* d_in, const int* in_sizes, int n_in,
                              void* d_out, int out_size, void* d_ws, size_t ws_size,
                              hipStream_t stream) {
  const int*   ui   = (const int*)d_in[0];
  const int*   ii   = (const int*)d_in[1];
  const float* utab = (const float*)d_in[2];
  const float* itab = (const float*)d_in[3];
  const float* Ws   = (const float*)d_in[4];   // [3,128,128]
  const float* bs   = (const float*)d_in[5];   // [3,128]
  const int B = in_sizes[0];                   // 200000 interactions
  const int n = 2 * B;                         // 400000 nodes

  float* X = (float*)d_out;                    // x lives in d_out throughout
  char*  ws = (char*)d_ws;
  float* H  = (float*)ws;                      // [n,128] = 204.8 MB
  size_t hb = (size_t)n * D * sizeof(float);
  int*   deg  = (int*)(ws + hb);               // [n] ints
  float* dinv = (float*)(ws + hb + (size_t)n * sizeof(int));  // [n] floats

  dim3 blk(256);
  const int tg = n * 32;                       // threads for row*float4 grids

  k_deg_init <<<(n + 255) / 256, blk, 0, stream>>>(deg, n);
  k_deg_count<<<(B + 255) / 256, blk, 0, stream>>>(ui, ii, deg, B);
  k_dinv     <<<(n + 255) / 256, blk, 0, stream>>>(deg, dinv, n);
  k_gather   <<<(tg + 255) / 256, blk, 0, stream>>>(ui, ii, utab, itab, X, B);

  for (int l = 0; l < 3; ++l) {
    k_gemm_wmma<<<n / 128, blk, 0, stream>>>(X, Ws + (size_t)l * D * D, H, n);
    k_zero     <<<(tg + 255) / 256, blk, 0, stream>>>((float4*)X, n * 32);
    k_scatter  <<<(B + 7) / 8, blk, 0, stream>>>(H, ui, ii, dinv, X, B);
    k_finish   <<<(tg + 255) / 256, blk, 0, stream>>>(X, H, dinv,
                                                      bs + (size_t)l * D, n);
  }
}